// Attention_3100966388365
// MI455X (gfx1250) — compile-verified
//
#include <hip/hip_runtime.h>
#include <hip/hip_bf16.h>

typedef __attribute__((ext_vector_type(2))) float v2f;
typedef __attribute__((ext_vector_type(8))) float v8f;

namespace {
constexpr int kBT = 8;           // B*T
constexpr int kS  = 576;         // H*W = 24*24
constexpr int kD  = 1024;
constexpr int kE  = 3 * kD;      // 3072
constexpr int kNH = 16;
constexpr int kHD = 64;
constexpr int kM  = kBT * kS;    // 4608 rows
constexpr int kQT = kS / 16;     // 36 key/query tiles
constexpr int kKVStride = 68;    // padded LDS row stride (68%64==4 -> no conflicts)
}

__device__ __forceinline__ v8f wmma_f32(v2f a, v2f b, v8f c) {
  // D = A(16x4,f32) * B(4x16,f32) + C(16x16,f32)
  return __builtin_amdgcn_wmma_f32_16x16x4_f32(false, a, false, b, (short)0, c,
                                               false, false);
}

__device__ __forceinline__ float half_max(float v) {
  v = fmaxf(v, __shfl_xor(v, 1, 32));
  v = fmaxf(v, __shfl_xor(v, 2, 32));
  v = fmaxf(v, __shfl_xor(v, 4, 32));
  v = fmaxf(v, __shfl_xor(v, 8, 32));
  return v;
}
__device__ __forceinline__ float half_sum(float v) {
  v += __shfl_xor(v, 1, 32);
  v += __shfl_xor(v, 2, 32);
  v += __shfl_xor(v, 4, 32);
  v += __shfl_xor(v, 8, 32);
  return v;
}

// Cooperatively stage a 16-row x 64-float (256B/row) tile into LDS with the
// gfx1250 async-to-LDS path (ASYNCcnt tracked). 128 threads x 16B x 2 issues.
__device__ __forceinline__ void stage_async(const float* gbase, unsigned lds_base,
                                            int tid) {
#pragma unroll
  for (int issue = 0; issue < 2; ++issue) {
    const int t   = issue * 128 + tid;
    const int row = t >> 4;        // 0..15
    const int seg = t & 15;        // 16B segment within the 256B row
    const float* src = gbase + (size_t)row * kE + seg * 4;
    const unsigned dst = lds_base + (unsigned)((row * kKVStride + seg * 4) * 4);
    asm volatile("global_load_async_to_lds_b128 %0, %1, off"
                 :: "v"(dst), "v"(src) : "memory");
  }
}

// ---------------------------------------------------------------------------
// NT GEMM: C[M,N] = A[M,kD] * B[N,kD]^T (+bias).
// One wave per 64x64 tile (4x4 register blocking: 16 WMMAs per 8 loads).
// ---------------------------------------------------------------------------
__global__ __launch_bounds__(256) void nt_gemm(const float* __restrict__ A,
                                               const float* __restrict__ Bm,
                                               const float* __restrict__ bias,
                                               float* __restrict__ C, int N) {
  const int lane = threadIdx.x & 31;
  const int gw   = blockIdx.x * 8 + (threadIdx.x >> 5);
  const int nt64 = N >> 6;
  const int tm   = gw / nt64;
  const int tn   = gw - tm * nt64;
  const int l15  = lane & 15;
  const int koff = (lane >> 4) << 1;   // K=0,1 for lanes 0-15; K=2,3 for 16-31

  const float* ap[4];
  const float* bp[4];
#pragma unroll
  for (int i = 0; i < 4; ++i) {
    ap[i] = A  + (size_t)(tm * 64 + i * 16 + l15) * kD + koff;
    bp[i] = Bm + (size_t)(tn * 64 + i * 16 + l15) * kD + koff;
  }

  v8f acc[4][4];
#pragma unroll
  for (int i = 0; i < 4; ++i)
#pragma unroll
    for (int j = 0; j < 4; ++j) acc[i][j] = v8f{};

  for (int k0 = 0; k0 < kD; k0 += 64) {
#pragma unroll
    for (int i = 0; i < 4; ++i) {     // prefetch next 64-deep k-panel
      __builtin_prefetch(ap[i] + k0 + 64, 0, 3);
      __builtin_prefetch(bp[i] + k0 + 64, 0, 3);
    }
#pragma unroll
    for (int kk = 0; kk < 64; kk += 4) {
      const int k = k0 + kk;
      v2f af[4], bf[4];
#pragma unroll
      for (int i = 0; i < 4; ++i) af[i] = *(const v2f*)(ap[i] + k);
#pragma unroll
      for (int j = 0; j < 4; ++j) bf[j] = *(const v2f*)(bp[j] + k);
#pragma unroll
      for (int i = 0; i < 4; ++i)
#pragma unroll
        for (int j = 0; j < 4; ++j)
          acc[i][j] = wmma_f32(af[i], bf[j], acc[i][j]);
    }
  }

  float bv[4];
#pragma unroll
  for (int j = 0; j < 4; ++j)
    bv[j] = bias ? bias[tn * 64 + j * 16 + l15] : 0.0f;

  const int rhalf = (lane >> 4) << 3;   // C layout: rows r / r+8
#pragma unroll
  for (int i = 0; i < 4; ++i)
#pragma unroll
    for (int r = 0; r < 8; ++r) {
      const int m = tm * 64 + i * 16 + rhalf + r;
      float* crow = C + (size_t)m * N + tn * 64 + l15;
#pragma unroll
      for (int j = 0; j < 4; ++j)
        crow[j * 16] = acc[i][j][r] + bv[j];
    }
}

// ---------------------------------------------------------------------------
// 2-D RoPE applied in place to the Q and K thirds of qkv[BT*S, 3072].
// ---------------------------------------------------------------------------
__global__ __launch_bounds__(256) void rope_kernel(float* __restrict__ QKV) {
  const int tid  = blockIdx.x * 256 + threadIdx.x;
  const int p    = tid & 31;
  const int head = (tid >> 5) & 15;
  const int qk   = (tid >> 9) & 1;
  const int u    = tid >> 10;           // bt*S + s
  const int s    = u % kS;
  const int hh   = s / 24;
  const int ww   = s - hh * 24;

  const float pos = (p < 16) ? (float)hh : (float)ww;
  const float ang = pos * __expf(-(float)(p & 15) * 0.57564627324851f); // ln(1e4)/16
  float sn, cs;
  __sincosf(ang, &sn, &cs);

  const size_t base = (size_t)u * kE + qk * kD + head * kHD + 2 * p;
  const float x0 = QKV[base], x1 = QKV[base + 1];
  QKV[base]     = x0 * cs - x1 * sn;
  QKV[base + 1] = x1 * cs + x0 * sn;
}

// ---------------------------------------------------------------------------
// Flash-style attention. Block = 4 waves, all on the same (batch,head), each
// wave owning one 16-row query tile. K/V 16-key tiles are double-buffered in
// LDS via async-to-LDS loads shared by the whole block.
// ---------------------------------------------------------------------------
__global__ __launch_bounds__(128) void attn_kernel(const float* __restrict__ QKV,
                                                   float* __restrict__ O) {
  __shared__ float kbuf[2][16 * kKVStride];
  __shared__ float vbuf[2][16 * kKVStride];
  __shared__ float ptile[4][16 * 17];

  const int tid  = threadIdx.x;
  const int lane = tid & 31;
  const int wid  = tid >> 5;
  const int bh   = blockIdx.x / 9;          // 128 (bt,head) pairs
  const int qt   = (blockIdx.x % 9) * 4 + wid;
  const int bt   = bh >> 4;
  const int h    = bh & 15;

  const float* base = QKV + (size_t)bt * kS * kE + h * kHD;
  const float* Qb = base;
  const float* Kb = base + kD;
  const float* Vb = base + 2 * kD;

  const int l15  = lane & 15;
  const int koff = (lane >> 4) << 1;

  // Preload this wave's 16x64 Q tile as 16 A-fragments.
  v2f qf[16];
  {
    const float* qp = Qb + (size_t)(qt * 16 + l15) * kE;
#pragma unroll
    for (int t = 0; t < 16; ++t) qf[t] = *(const v2f*)(qp + t * 4 + koff);
  }

  float mrow[8], lrow[8];
#pragma unroll
  for (int r = 0; r < 8; ++r) { mrow[r] = -3.0e38f; lrow[r] = 0.0f; }
  v8f oacc[4] = {v8f{}, v8f{}, v8f{}, v8f{}};

  float* lp = &ptile[wid][0];
  const float scale = 0.125f;   // 64^-0.5

  // Stage kt=0 into buffer 0.
  stage_async(Kb, (unsigned)(uintptr_t)&kbuf[0][0], tid);
  stage_async(Vb, (unsigned)(uintptr_t)&vbuf[0][0], tid);

  for (int kt = 0; kt < kQT; ++kt) {
    const int cur = kt & 1;
    asm volatile("s_wait_asynccnt 0x0" ::: "memory");
    __syncthreads();   // stage(kt) visible block-wide; compute(kt-1) finished
    if (kt + 1 < kQT) {
      const size_t nxt = (size_t)(kt + 1) * 16 * kE;
      stage_async(Kb + nxt, (unsigned)(uintptr_t)&kbuf[cur ^ 1][0], tid);
      stage_async(Vb + nxt, (unsigned)(uintptr_t)&vbuf[cur ^ 1][0], tid);
    }
    const float* kls = &kbuf[cur][0];
    const float* vls = &vbuf[cur][0];

    // S = Q * K^T (B-fragments from LDS; key index = l15)
    v8f sc = {};
#pragma unroll
    for (int t = 0; t < 16; ++t) {
      v2f bf = *(const v2f*)(kls + l15 * kKVStride + t * 4 + koff);
      sc = wmma_f32(qf[t], bf, sc);
    }
    // Online softmax per C-layout row (r -> rows r / r+8).
#pragma unroll
    for (int r = 0; r < 8; ++r) {
      float v  = sc[r] * scale;
      float mn = fmaxf(mrow[r], half_max(v));
      float alpha = __expf(mrow[r] - mn);
      float p = __expf(v - mn);
      mrow[r] = mn;
      lrow[r] = lrow[r] * alpha + half_sum(p);
#pragma unroll
      for (int c = 0; c < 4; ++c) oacc[c][r] *= alpha;
      lp[(r + ((lane >> 4) << 3)) * 17 + l15] = p;   // C-layout -> LDS (m,n)
    }
    // O += P * V (P back in A-layout from per-wave LDS tile; V from LDS).
#pragma unroll
    for (int t = 0; t < 4; ++t) {
      v2f a;
      a.x = lp[l15 * 17 + t * 4 + koff];
      a.y = lp[l15 * 17 + t * 4 + koff + 1];
#pragma unroll
      for (int c = 0; c < 4; ++c) {
        v2f b;
        b.x = vls[(t * 4 + koff) * kKVStride + c * 16 + l15];
        b.y = vls[(t * 4 + koff + 1) * kKVStride + c * 16 + l15];
        oacc[c] = wmma_f32(a, b, oacc[c]);
      }
    }
  }

  // Normalize and write [BT*S, D].
  const int m0 = qt * 16 + ((lane >> 4) << 3);
#pragma unroll
  for (int r = 0; r < 8; ++r) {
    const float inv = 1.0f / lrow[r];
    const size_t rowoff = ((size_t)bt * kS + m0 + r) * kD + h * kHD;
#pragma unroll
    for (int c = 0; c < 4; ++c)
      O[rowoff + c * 16 + l15] = oacc[c][r] * inv;
  }
}

// ---------------------------------------------------------------------------
extern "C" void kernel_launch(void* const* d_in, const int* in_sizes, int n_in,
                              void* d_out, int out_size, void* d_ws, size_t ws_size,
                              hipStream_t stream) {
  const float* x     = (const float*)d_in[0];   // [8,576,1024]
  const float* w_qkv = (const float*)d_in[1];   // [3072,1024]
  const float* w_out = (const float*)d_in[2];   // [1024,1024]
  const float* b_out = (const float*)d_in[3];   // [1024]
  float* out = (float*)d_out;                   // [8,576,1024]

  float* qkv = (float*)d_ws;                    // 8*576*3072 f32 = 56.6 MB
  float* o   = qkv + (size_t)kM * kE;           // 8*576*1024 f32 = 18.9 MB

  // QKV GEMM: (72 x 48) 64x64 tiles, 8 waves/block -> 432 blocks
  nt_gemm<<<(kM / 64) * (kE / 64) / 8, 256, 0, stream>>>(x, w_qkv, nullptr, qkv, kE);

  // RoPE: BT*S*2*NH*32 pair-threads = 4,718,592 -> 18432 blocks
  rope_kernel<<<(kBT * kS * 2 * kNH * 32) / 256, 256, 0, stream>>>(qkv);

  // Attention: 128 (bt,head) x 9 q-tile groups -> 1152 blocks of 4 waves
  attn_kernel<<<kBT * kNH * (kQT / 4), 128, 0, stream>>>(qkv, o);

  // Output projection + bias: (72 x 16) 64x64 tiles -> 144 blocks
  nt_gemm<<<(kM / 64) * (kD / 64) / 8, 256, 0, stream>>>(o, w_out, b_out, out, kD);
}